// PointNet2GraspQNet_61409442399003
// MI455X (gfx1250) — compile-verified
//
#include <hip/hip_runtime.h>
#include <math.h>

typedef __attribute__((ext_vector_type(2))) float v2f;
typedef __attribute__((ext_vector_type(8))) float v8f;

#define BN_EPS 1e-5f
#define MT 4   // 16-row tiles per wave (64 rows)

// ======================================================================
// GEMM: Y[r,o] = sum_c X[r,c] * W[o,c] (+ bias[o])   (einsum '...c,oc->...o')
// One wave per 64(M)x16(N) block: 4 accumulators share one B fragment per
// K-step of V_WMMA_F32_16X16X4_F32.
// A (16x4 MxK): VGPR0: lanes0-15 K=k0, lanes16-31 K=k0+2; VGPR1: K=k0+1/k0+3
// B (4x16 KxN): B[k][n] = W[n][k]; same half-lane K split as A.
// C/D (16x16): VGPR v: lanes0-15 M=v, lanes16-31 M=v+8; N = lane&15.
// ======================================================================
template <bool VEC>
__global__ void gemm_wmma_f32(const float* __restrict__ X, const float* __restrict__ W,
                              const float* __restrict__ bias, float* __restrict__ Y,
                              int R, int C, int O) {
  const int wave   = (int)((blockIdx.x * blockDim.x + threadIdx.x) >> 5);
  const int lane   = (int)(threadIdx.x & 31);
  const int tilesN = (O + 15) >> 4;
  const int tilesM = (R + 16 * MT - 1) / (16 * MT);
  if (wave >= tilesM * tilesN) return;            // wave-uniform: EXEC stays full
  const int tm   = wave / tilesN;
  const int tn   = wave - tm * tilesN;
  const int row0 = tm * (16 * MT);
  const int col0 = tn << 4;
  const int half = lane >> 4;
  const int l15  = lane & 15;
  const int Cal  = C & ~3;

  v8f acc[MT];
  {
    v8f z = {};
#pragma unroll
    for (int t = 0; t < MT; ++t) acc[t] = z;
  }

  if (row0 + 16 * MT <= R && col0 + 16 <= O) {
    // ---------- fast path: fully in-bounds block, unconditional loads ----------
    const int n = col0 + l15;
    const float* Wp  = W + (long long)n * C;
    const float* Xp0 = X + (long long)(row0 + l15) * C;
    const long long rstride = (long long)16 * C;

    int k0 = 0;
    for (; k0 < Cal; k0 += 4) {
      const int ka = k0 + (half << 1);
      v2f Bv;
      if (VEC) {
        Bv = *(const v2f*)(Wp + ka);
      } else {
        Bv = (v2f){Wp[ka], Wp[ka + 1]};
      }
#pragma unroll
      for (int t = 0; t < MT; ++t) {
        const float* Xp = Xp0 + rstride * t;
        v2f Av;
        if (VEC) {
          Av = *(const v2f*)(Xp + ka);
        } else {
          Av = (v2f){Xp[ka], Xp[ka + 1]};
        }
        acc[t] = __builtin_amdgcn_wmma_f32_16x16x4_f32(false, Av, false, Bv, (short)0,
                                                       acc[t], false, false);
      }
    }
    if (k0 < C) {   // K tail (C in {3,131,259}); clamp+scale keeps loads branchless
      const int ka  = k0 + (half << 1);
      const int k0c = (ka     < C) ? ka     : (C - 1);
      const int k1c = (ka + 1 < C) ? ka + 1 : (C - 1);
      const float s0 = (ka     < C) ? 1.f : 0.f;
      const float s1 = (ka + 1 < C) ? 1.f : 0.f;
      v2f Bv = {Wp[k0c] * s0, Wp[k1c] * s1};
#pragma unroll
      for (int t = 0; t < MT; ++t) {
        const float* Xp = Xp0 + rstride * t;
        v2f Av = {Xp[k0c] * s0, Xp[k1c] * s1};
        acc[t] = __builtin_amdgcn_wmma_f32_16x16x4_f32(false, Av, false, Bv, (short)0,
                                                       acc[t], false, false);
      }
    }
    const float bb = bias ? bias[n] : 0.f;
#pragma unroll
    for (int t = 0; t < MT; ++t) {
#pragma unroll
      for (int v = 0; v < 8; ++v) {
        const int m = row0 + t * 16 + v + (half << 3);
        Y[(long long)m * O + n] = acc[t][v] + bb;
      }
    }
  } else {
    // ---------- slow path (head layers: R=8 and/or O=1): clamp+scale guards ----
    const int n   = col0 + l15;
    const int nc  = (n < O) ? n : (O - 1);
    const float sb = (n < O) ? 1.f : 0.f;
    const float* Wp = W + (long long)nc * C;
    for (int t = 0; t < MT; ++t) {
      const int r0 = row0 + t * 16;
      if (r0 >= R) break;
      const int arow = r0 + l15;
      const int ac   = (arow < R) ? arow : (R - 1);
      const float sa = (arow < R) ? 1.f : 0.f;
      const float* Xp = X + (long long)ac * C;
      for (int k0 = 0; k0 < C; k0 += 4) {
        const int ka  = k0 + (half << 1);
        const int k0c = (ka     < C) ? ka     : (C - 1);
        const int k1c = (ka + 1 < C) ? ka + 1 : (C - 1);
        const float s0 = (ka     < C) ? 1.f : 0.f;
        const float s1 = (ka + 1 < C) ? 1.f : 0.f;
        v2f Av = {Xp[k0c] * (s0 * sa), Xp[k1c] * (s1 * sa)};
        v2f Bv = {Wp[k0c] * (s0 * sb), Wp[k1c] * (s1 * sb)};
        acc[t] = __builtin_amdgcn_wmma_f32_16x16x4_f32(false, Av, false, Bv, (short)0,
                                                       acc[t], false, false);
      }
      if (n < O) {
        const float bb = bias ? bias[n] : 0.f;
#pragma unroll
        for (int v = 0; v < 8; ++v) {
          const int m = r0 + v + (half << 3);
          if (m < R) Y[(long long)m * O + n] = acc[t][v] + bb;
        }
      }
    }
  }
}

// ======================================================================
// BatchNorm (training mode): per-channel mean / rsqrt(var+eps) over R rows
// ======================================================================
__global__ void bn_stats(const float* __restrict__ Y, float* __restrict__ mean,
                         float* __restrict__ invstd, int R, int O) {
  __shared__ float sh_s[256];
  __shared__ float sh_q[256];
  const int o = blockIdx.x;
  float s = 0.f, q = 0.f;
  for (int r = threadIdx.x; r < R; r += 256) {
    float v = Y[(long long)r * O + o];
    s += v;
    q += v * v;
  }
  sh_s[threadIdx.x] = s;
  sh_q[threadIdx.x] = q;
  __syncthreads();
  for (int st = 128; st > 0; st >>= 1) {
    if ((int)threadIdx.x < st) {
      sh_s[threadIdx.x] += sh_s[threadIdx.x + st];
      sh_q[threadIdx.x] += sh_q[threadIdx.x + st];
    }
    __syncthreads();
  }
  if (threadIdx.x == 0) {
    float m   = sh_s[0] / (float)R;
    float var = sh_q[0] / (float)R - m * m;
    mean[o]   = m;
    invstd[o] = rsqrtf(var + BN_EPS);
  }
}

__global__ void bn_apply(float* __restrict__ Y, const float* __restrict__ mean,
                         const float* __restrict__ invstd, int R, int O, int doRelu) {
  long long i = (long long)blockIdx.x * blockDim.x + threadIdx.x;
  if (i >= (long long)R * O) return;
  int o = (int)(i % O);
  float v = (Y[i] - mean[o]) * invstd[o];
  Y[i] = doRelu ? fmaxf(v, 0.f) : v;
}

__global__ void relu_k(float* __restrict__ Y, int n) {
  int i = (int)(blockIdx.x * blockDim.x + threadIdx.x);
  if (i < n) Y[i] = fmaxf(Y[i], 0.f);
}

// ======================================================================
// Farthest point sampling: one 256-thread workgroup per batch.
// dd[b,:] scratch holds running min-dist; each thread owns a strided slice.
// ======================================================================
__global__ void fps_kernel(const float* __restrict__ xyz, int N, int npoint,
                           int* __restrict__ fidx, float* __restrict__ dists) {
  const int b = blockIdx.x;
  const float* p = xyz + (long long)b * N * 3;
  float* dd = dists + (long long)b * N;
  __shared__ float sh_best[256];
  __shared__ int   sh_bi[256];
  __shared__ int   sh_far;
  for (int i = threadIdx.x; i < N; i += 256) dd[i] = 1e10f;
  if (threadIdx.x == 0) sh_far = 0;
  __syncthreads();
  for (int it = 0; it < npoint; ++it) {
    const int far = sh_far;
    if (threadIdx.x == 0) fidx[(long long)b * npoint + it] = far;
    const float fx = p[far * 3 + 0];
    const float fy = p[far * 3 + 1];
    const float fz = p[far * 3 + 2];
    float best = -1.f;
    int   bi   = 0;
    for (int i = threadIdx.x; i < N; i += 256) {
      float dx = p[i * 3 + 0] - fx;
      float dy = p[i * 3 + 1] - fy;
      float dz = p[i * 3 + 2] - fz;
      float d  = dx * dx + dy * dy + dz * dz;
      float nv = fminf(dd[i], d);
      dd[i] = nv;
      if (nv > best) { best = nv; bi = i; }    // strict > keeps lowest index
    }
    sh_best[threadIdx.x] = best;
    sh_bi[threadIdx.x]   = bi;
    __syncthreads();
    for (int st = 128; st > 0; st >>= 1) {
      if ((int)threadIdx.x < st) {
        float ob = sh_best[threadIdx.x + st];
        int   oi = sh_bi[threadIdx.x + st];
        if (ob > sh_best[threadIdx.x] ||
            (ob == sh_best[threadIdx.x] && oi < sh_bi[threadIdx.x])) {
          sh_best[threadIdx.x] = ob;
          sh_bi[threadIdx.x]   = oi;
        }
      }
      __syncthreads();
    }
    if (threadIdx.x == 0) sh_far = sh_bi[0];
    __syncthreads();
  }
}

// ======================================================================
// Gather FPS centers: new_xyz[b,s,:] = xyz[b, fidx[b,s], :]
// ======================================================================
__global__ void gather_centers(const float* __restrict__ xyz, const int* __restrict__ fidx,
                               float* __restrict__ out, int B, int N, int S) {
  int i = (int)(blockIdx.x * blockDim.x + threadIdx.x);   // over B*S*3
  if (i >= B * S * 3) return;
  int c  = i % 3;
  int bs = i / 3;
  int b  = bs / S;
  out[i] = xyz[((long long)b * N + fidx[bs]) * 3 + c];
}

// ======================================================================
// Ball query: one wave32 per center; ordered first-nsample selection via
// ballot + prefix popcount (wave32-native equivalent of the rank/top_k trick)
// ======================================================================
__global__ void ball_query_kernel(const float* __restrict__ centers,
                                  const float* __restrict__ xyz,
                                  int B, int S, int N, float radius, int nsample,
                                  int* __restrict__ idx) {
  const int gw   = (int)((blockIdx.x * blockDim.x + threadIdx.x) >> 5);
  const int lane = (int)(threadIdx.x & 31);
  if (gw >= B * S) return;                 // wave-uniform
  const int b = gw / S;
  const float cx = centers[gw * 3 + 0];
  const float cy = centers[gw * 3 + 1];
  const float cz = centers[gw * 3 + 2];
  const float* p = xyz + (long long)b * N * 3;
  int* out = idx + (long long)gw * nsample;
  const float r2 = radius * radius;
  int count = 0;
  int firstIdx = -1;
  for (int j0 = 0; j0 < N && count < nsample; j0 += 32) {
    const int j = j0 + lane;
    bool pred = false;
    if (j < N) {
      float dx = p[j * 3 + 0] - cx;
      float dy = p[j * 3 + 1] - cy;
      float dz = p[j * 3 + 2] - cz;
      pred = (dx * dx + dy * dy + dz * dz) < r2;
    }
    unsigned mask = __builtin_amdgcn_ballot_w32(pred);
    if (mask && firstIdx < 0) firstIdx = j0 + (int)__builtin_ctz(mask); // wave-uniform
    int prefix = __builtin_popcount(mask & ((lane == 31) ? 0x7fffffffu
                                                         : ((1u << lane) - 1u)));
    if (pred) {
      int slot = count + prefix;
      if (slot < nsample) out[slot] = j;
    }
    count += __builtin_popcount(mask);
  }
  if (firstIdx < 0) firstIdx = 0;          // unreachable: center is its own neighbor
  int start = count < nsample ? count : nsample;
  for (int t = start + lane; t < nsample; t += 32) out[t] = firstIdx;
}

// ======================================================================
// Group: out[b,s,j, 0:3]   = (xyz[b,idx,:] - center)/radius
//        out[b,s,j, 3:3+Cf] = feats[b,idx,:]
// ======================================================================
__global__ void group_points(const float* __restrict__ xyz, const float* __restrict__ centers,
                             const float* __restrict__ feats, const int* __restrict__ idx,
                             float* __restrict__ out, int B, int N, int S, int ns, int Cf,
                             float inv_radius) {
  const int Cin = 3 + Cf;
  const long long total = (long long)B * S * ns * Cin;
  long long i = (long long)blockIdx.x * blockDim.x + threadIdx.x;
  if (i >= total) return;
  int c = (int)(i % Cin);
  long long g = i / Cin;               // b*S*ns flat group index
  int bs = (int)(g / ns);
  int b  = bs / S;
  int j  = idx[g];
  float v;
  if (c < 3) {
    v = (xyz[((long long)b * N + j) * 3 + c] - centers[(long long)bs * 3 + c]) * inv_radius;
  } else {
    v = feats[((long long)b * N + j) * Cf + (c - 3)];
  }
  out[i] = v;
}

// ======================================================================
// Maxpool over middle dim: X[G,ns,C] -> Y[G,C]
// ======================================================================
__global__ void maxpool_k(const float* __restrict__ X, float* __restrict__ Y,
                          int G, int ns, int C) {
  long long i = (long long)blockIdx.x * blockDim.x + threadIdx.x;
  if (i >= (long long)G * C) return;
  int g = (int)(i / C);
  int c = (int)(i % C);
  const float* p = X + (long long)g * ns * C + c;
  float m = p[0];
  for (int j = 1; j < ns; ++j) m = fmaxf(m, p[(long long)j * C]);
  Y[i] = m;
}

// ======================================================================
// concat2: out[r,:] = [A[r,0:C1], B[r,0:C2]]
// ======================================================================
__global__ void concat2(const float* __restrict__ A, const float* __restrict__ Bm,
                        float* __restrict__ out, int R, int C1, int C2) {
  int C = C1 + C2;
  long long i = (long long)blockIdx.x * blockDim.x + threadIdx.x;
  if (i >= (long long)R * C) return;
  int r = (int)(i / C);
  int c = (int)(i % C);
  out[i] = (c < C1) ? A[(long long)r * C1 + c] : Bm[(long long)r * C2 + (c - C1)];
}

// ======================================================================
// Host-side orchestration
// ======================================================================
static inline unsigned cdiv_u(long long a, long long b) { return (unsigned)((a + b - 1) / b); }

static void run_gemm(const float* X, const float* W, const float* bias, float* Y,
                     int R, int C, int O, hipStream_t s) {
  long long waves = (long long)((R + 16 * MT - 1) / (16 * MT)) * ((O + 15) / 16);
  unsigned blocks = cdiv_u(waves, 8);   // 8 waves (256 threads) per block
  if ((C & 3) == 0) {
    gemm_wmma_f32<true><<<blocks, 256, 0, s>>>(X, W, bias, Y, R, C, O);
  } else {
    gemm_wmma_f32<false><<<blocks, 256, 0, s>>>(X, W, bias, Y, R, C, O);
  }
}

static void run_bn(float* Y, float* mean, float* invstd, int R, int O, int doRelu,
                   hipStream_t s) {
  bn_stats<<<O, 256, 0, s>>>(Y, mean, invstd, R, O);
  bn_apply<<<cdiv_u((long long)R * O, 256), 256, 0, s>>>(Y, mean, invstd, R, O, doRelu);
}

extern "C" void kernel_launch(void* const* d_in, const int* in_sizes, int n_in,
                              void* d_out, int out_size, void* d_ws, size_t ws_size,
                              hipStream_t stream) {
  (void)in_sizes; (void)n_in; (void)out_size; (void)ws_size;
  const float* obj     = (const float*)d_in[0];   // [8,20000,3]
  const float* grip    = (const float*)d_in[1];   // [8,512,3]
  const float* sa1w0   = (const float*)d_in[2];   // [64,3]
  const float* sa1w1   = (const float*)d_in[3];   // [64,64]
  const float* sa1w2   = (const float*)d_in[4];   // [128,64]
  const float* sa2w0   = (const float*)d_in[5];   // [128,131]
  const float* sa2w1   = (const float*)d_in[6];   // [128,128]
  const float* sa2w2   = (const float*)d_in[7];   // [256,128]
  const float* sa3w0   = (const float*)d_in[8];   // [256,259]
  const float* sa3w1   = (const float*)d_in[9];   // [512,256]
  const float* sa3w2   = (const float*)d_in[10];  // [1024,512]
  const float* gew0    = (const float*)d_in[11];  // [64,3]
  const float* gew1    = (const float*)d_in[12];  // [128,64]
  const float* gew2    = (const float*)d_in[13];  // [128,128]
  const float* hw0     = (const float*)d_in[14];  // [512,1152]
  const float* hb0     = (const float*)d_in[15];  // [512]
  const float* hw1     = (const float*)d_in[16];  // [256,512]
  const float* hb1     = (const float*)d_in[17];  // [256]
  const float* hw2     = (const float*)d_in[18];  // [1,256]
  const float* hb2     = (const float*)d_in[19];  // [1]
  float* out = (float*)d_out;                     // [8,1]

  const int B = 8, N1 = 20000, S1 = 512, NS1 = 32;
  const int S2 = 128, NS2 = 64;

  // ---- workspace bump allocator ----
  char* wsp = (char*)d_ws;
  auto alloc = [&](size_t bytes) -> void* {
    void* p = (void*)wsp;
    wsp += (bytes + 255) & ~(size_t)255;
    return p;
  };
  float* bufA    = (float*)alloc((size_t)131072 * 128 * 4);   // ping
  float* bufB    = (float*)alloc((size_t)131072 * 128 * 4);   // pong
  float* dists   = (float*)alloc((size_t)B * N1 * 4);
  int*   fidx1   = (int*)  alloc((size_t)B * S1 * 4);
  int*   idx1    = (int*)  alloc((size_t)B * S1 * NS1 * 4);
  int*   fidx2   = (int*)  alloc((size_t)B * S2 * 4);
  int*   idx2    = (int*)  alloc((size_t)B * S2 * NS2 * 4);
  float* nxyz1   = (float*)alloc((size_t)B * S1 * 3 * 4);
  float* nxyz2   = (float*)alloc((size_t)B * S2 * 3 * 4);
  float* f1      = (float*)alloc((size_t)B * S1 * 128 * 4);
  float* f2      = (float*)alloc((size_t)B * S2 * 256 * 4);
  float* gfeat   = (float*)alloc((size_t)B * 128 * 4);
  float* ofeat   = (float*)alloc((size_t)B * 1024 * 4);
  float* comb    = (float*)alloc((size_t)B * 1152 * 4);
  float* h0      = (float*)alloc((size_t)B * 512 * 4);
  float* h1      = (float*)alloc((size_t)B * 256 * 4);
  float* mean    = (float*)alloc((size_t)2048 * 4);
  float* invstd  = (float*)alloc((size_t)2048 * 4);

  // ================= gripper encoder (rows = 8*512 = 4096) =================
  run_gemm(grip, gew0, nullptr, bufA, 4096, 3, 64, stream);
  run_bn(bufA, mean, invstd, 4096, 64, 1, stream);
  run_gemm(bufA, gew1, nullptr, bufB, 4096, 64, 128, stream);
  run_bn(bufB, mean, invstd, 4096, 128, 1, stream);
  run_gemm(bufB, gew2, nullptr, bufA, 4096, 128, 128, stream);
  run_bn(bufA, mean, invstd, 4096, 128, 1, stream);
  maxpool_k<<<cdiv_u((long long)B * 128, 256), 256, 0, stream>>>(bufA, gfeat, B, 512, 128);

  // ================= SA1: 20000 -> 512 centers, r=0.2, ns=32 ================
  fps_kernel<<<B, 256, 0, stream>>>(obj, N1, S1, fidx1, dists);
  gather_centers<<<cdiv_u((long long)B * S1 * 3, 256), 256, 0, stream>>>(obj, fidx1, nxyz1,
                                                                         B, N1, S1);
  ball_query_kernel<<<cdiv_u((long long)B * S1 * 32, 256), 256, 0, stream>>>(
      nxyz1, obj, B, S1, N1, 0.2f, NS1, idx1);
  {
    long long tot = (long long)B * S1 * NS1 * 3;
    group_points<<<cdiv_u(tot, 256), 256, 0, stream>>>(obj, nxyz1, nullptr, idx1, bufA,
                                                       B, N1, S1, NS1, 0, 1.f / 0.2f);
  }
  const int R1 = B * S1 * NS1;   // 131072
  run_gemm(bufA, sa1w0, nullptr, bufB, R1, 3, 64, stream);
  run_bn(bufB, mean, invstd, R1, 64, 1, stream);
  run_gemm(bufB, sa1w1, nullptr, bufA, R1, 64, 64, stream);
  run_bn(bufA, mean, invstd, R1, 64, 1, stream);
  run_gemm(bufA, sa1w2, nullptr, bufB, R1, 64, 128, stream);
  run_bn(bufB, mean, invstd, R1, 128, 1, stream);
  maxpool_k<<<cdiv_u((long long)B * S1 * 128, 256), 256, 0, stream>>>(bufB, f1, B * S1,
                                                                      NS1, 128);

  // ================= SA2: 512 -> 128 centers, r=0.4, ns=64 ==================
  fps_kernel<<<B, 256, 0, stream>>>(nxyz1, S1, S2, fidx2, dists);
  gather_centers<<<cdiv_u((long long)B * S2 * 3, 256), 256, 0, stream>>>(nxyz1, fidx2, nxyz2,
                                                                         B, S1, S2);
  ball_query_kernel<<<cdiv_u((long long)B * S2 * 32, 256), 256, 0, stream>>>(
      nxyz2, nxyz1, B, S2, S1, 0.4f, NS2, idx2);
  {
    long long tot = (long long)B * S2 * NS2 * 131;
    group_points<<<cdiv_u(tot, 256), 256, 0, stream>>>(nxyz1, nxyz2, f1, idx2, bufA,
                                                       B, S1, S2, NS2, 128, 1.f / 0.4f);
  }
  const int R2 = B * S2 * NS2;   // 65536
  run_gemm(bufA, sa2w0, nullptr, bufB, R2, 131, 128, stream);
  run_bn(bufB, mean, invstd, R2, 128, 1, stream);
  run_gemm(bufB, sa2w1, nullptr, bufA, R2, 128, 128, stream);
  run_bn(bufA, mean, invstd, R2, 128, 1, stream);
  run_gemm(bufA, sa2w2, nullptr, bufB, R2, 128, 256, stream);
  run_bn(bufB, mean, invstd, R2, 256, 1, stream);
  maxpool_k<<<cdiv_u((long long)B * S2 * 256, 256), 256, 0, stream>>>(bufB, f2, B * S2,
                                                                      NS2, 256);

  // ================= SA3: group_all, rows = 8*128 = 1024 ====================
  const int R3 = B * S2;         // 1024
  concat2<<<cdiv_u((long long)R3 * 259, 256), 256, 0, stream>>>(nxyz2, f2, bufA, R3, 3, 256);
  run_gemm(bufA, sa3w0, nullptr, bufB, R3, 259, 256, stream);
  run_bn(bufB, mean, invstd, R3, 256, 1, stream);
  run_gemm(bufB, sa3w1, nullptr, bufA, R3, 256, 512, stream);
  run_bn(bufA, mean, invstd, R3, 512, 1, stream);
  run_gemm(bufA, sa3w2, nullptr, bufB, R3, 512, 1024, stream);
  run_bn(bufB, mean, invstd, R3, 1024, 1, stream);
  maxpool_k<<<cdiv_u((long long)B * 1024, 256), 256, 0, stream>>>(bufB, ofeat, B, S2, 1024);

  // ================= head ===================================================
  concat2<<<cdiv_u((long long)B * 1152, 256), 256, 0, stream>>>(gfeat, ofeat, comb, B, 128,
                                                                1024);
  run_bn(comb, mean, invstd, B, 1152, 0, stream);              // _bn(comb), no relu
  run_gemm(comb, hw0, hb0, h0, B, 1152, 512, stream);
  run_bn(h0, mean, invstd, B, 512, 1, stream);
  run_gemm(h0, hw1, hb1, h1, B, 512, 256, stream);
  run_bn(h1, mean, invstd, B, 256, 1, stream);
  run_gemm(h1, hw2, hb2, out, B, 256, 1, stream);
  relu_k<<<1, 32, 0, stream>>>(out, B);
}